// Encoder_14130442404379
// MI455X (gfx1250) — compile-verified
//
#include <hip/hip_runtime.h>
#include <math.h>

// ---------------------------------------------------------------------------
// Bidirectional Mamba encoder for MI455X (gfx1250, wave32).
// GEMMs use V_WMMA_F32_16X16X4_F32 with clamped (branch-free) fragment loads
// and software-pipelined K loop; scan parallel over B*di channels.
// ---------------------------------------------------------------------------

typedef float v2f __attribute__((ext_vector_type(2)));
typedef float v8f __attribute__((ext_vector_type(8)));

#define BATCH   16
#define SEQ     256
#define DMODEL  256
#define DI      512         // 2*d
#define DSTATE  16
#define DTR     16
#define XDBL_LD 48          // dtr + 2*ds
#define MROWS   (BATCH*SEQ) // 4096

__device__ __forceinline__ float silu_f(float v) { return v / (1.f + __expf(-v)); }
__device__ __forceinline__ int clamp_i(int v, int hi) { return v < hi ? v : hi; }

// ---------------------------------------------------------------------------
// C[M,N] = X[M,K] @ W[N,K]^T   (all fp32 row-major; X row stride lda,
// W row stride ldw, C row stride ldc).  Block = 256 threads = 8 waves laid
// out 2(M) x 4(N); each wave computes a 32x32 C region via 2x2 WMMA tiles.
// Macro tile per block: 64(M) x 128(N).  K must be a multiple of 4, >= 4.
//
// Out-of-range rows/cols are handled by CLAMPING the load index (the loaded
// garbage only feeds accumulator lanes that are never stored), so every
// fragment load is an unconditional global_load_b64 and EXEC stays all-ones
// around the WMMAs.  The K loop is double-buffered so each iteration's WMMAs
// consume the previous iteration's loads.
// ---------------------------------------------------------------------------
__global__ __launch_bounds__(256)
void wmma_gemm_xwt(const float* __restrict__ X, int lda,
                   const float* __restrict__ W, int ldw,
                   float* __restrict__ C, int ldc,
                   int M, int N, int K) {
  const int lane = threadIdx.x & 31;
  const int wave = threadIdx.x >> 5;
  const int lr   = lane & 15;   // lane row within half-wave
  const int lh   = lane >> 4;   // half select (K phase for A/B frags)
  const int m0 = blockIdx.y * 64  + (wave >> 2) * 32;
  const int n0 = blockIdx.x * 128 + (wave & 3) * 32;

  v8f acc00 = {}; v8f acc01 = {}; v8f acc10 = {}; v8f acc11 = {};

  // A fragment (16x4 f32): lanes 0-15 -> K=k,k+1 ; lanes 16-31 -> K=k+2,k+3
  const int ar0 = clamp_i(m0 + lr,      M - 1);
  const int ar1 = clamp_i(m0 + 16 + lr, M - 1);
  const int bn0 = clamp_i(n0 + lr,      N - 1);
  const int bn1 = clamp_i(n0 + 16 + lr, N - 1);
  const float* Xp0 = X + (size_t)ar0 * lda + lh * 2;
  const float* Xp1 = X + (size_t)ar1 * lda + lh * 2;
  const float* Wp0 = W + (size_t)bn0 * ldw + lh * 2;
  const float* Wp1 = W + (size_t)bn1 * ldw + lh * 2;

  // Prologue: fragments for k = 0.
  v2f a0 = *(const v2f*)(Xp0);
  v2f a1 = *(const v2f*)(Xp1);
  v2f b0 = *(const v2f*)(Wp0);
  v2f b1 = *(const v2f*)(Wp1);

  for (int k = 4; k < K; k += 4) {
    // Issue next iteration's loads first; WMMAs below cover their latency.
    v2f na0 = *(const v2f*)(Xp0 + k);
    v2f na1 = *(const v2f*)(Xp1 + k);
    v2f nb0 = *(const v2f*)(Wp0 + k);
    v2f nb1 = *(const v2f*)(Wp1 + k);
    acc00 = __builtin_amdgcn_wmma_f32_16x16x4_f32(false, a0, false, b0, (short)0, acc00, false, false);
    acc01 = __builtin_amdgcn_wmma_f32_16x16x4_f32(false, a0, false, b1, (short)0, acc01, false, false);
    acc10 = __builtin_amdgcn_wmma_f32_16x16x4_f32(false, a1, false, b0, (short)0, acc10, false, false);
    acc11 = __builtin_amdgcn_wmma_f32_16x16x4_f32(false, a1, false, b1, (short)0, acc11, false, false);
    a0 = na0; a1 = na1; b0 = nb0; b1 = nb1;
  }
  // Epilogue: last k-step.
  acc00 = __builtin_amdgcn_wmma_f32_16x16x4_f32(false, a0, false, b0, (short)0, acc00, false, false);
  acc01 = __builtin_amdgcn_wmma_f32_16x16x4_f32(false, a0, false, b1, (short)0, acc01, false, false);
  acc10 = __builtin_amdgcn_wmma_f32_16x16x4_f32(false, a1, false, b0, (short)0, acc10, false, false);
  acc11 = __builtin_amdgcn_wmma_f32_16x16x4_f32(false, a1, false, b1, (short)0, acc11, false, false);

  // C/D layout (16x16 f32): vgpr r -> row r + 8*lh, col = lane&15
  #pragma unroll
  for (int r = 0; r < 8; ++r) {
    const int row0 = m0 + r + lh * 8;
    const int row1 = m0 + 16 + r + lh * 8;
    const int col0 = n0 + lr;
    const int col1 = n0 + 16 + lr;
    if (row0 < M && col0 < N) C[(size_t)row0 * ldc + col0] = acc00[r];
    if (row0 < M && col1 < N) C[(size_t)row0 * ldc + col1] = acc01[r];
    if (row1 < M && col0 < N) C[(size_t)row1 * ldc + col0] = acc10[r];
    if (row1 < M && col1 < N) C[(size_t)row1 * ldc + col1] = acc11[r];
  }
}

// ---------------------------------------------------------------------------
// Flip along the sequence axis: out[b,l,:] = in[b,L-1-l,:]
// ---------------------------------------------------------------------------
__global__ __launch_bounds__(256)
void flip_seq(const float* __restrict__ in, float* __restrict__ out,
              int B, int L, int d) {
  int i = blockIdx.x * blockDim.x + threadIdx.x;
  int total = B * L * d;
  if (i >= total) return;
  int c = i % d;
  int l = (i / d) % L;
  int b = i / (d * L);
  out[i] = in[((size_t)b * L + (L - 1 - l)) * d + c];
}

// ---------------------------------------------------------------------------
// Depthwise causal conv (width K, left zero-pad) + bias + SiLU on xz[...,:di]
// xz row stride = 2*di.  Output xs_act (row stride di).
// ---------------------------------------------------------------------------
__global__ __launch_bounds__(256)
void conv_silu(const float* __restrict__ xz,
               const float* __restrict__ conv_w, const float* __restrict__ conv_b,
               float* __restrict__ xs_act, int B, int L, int di, int K) {
  int i = blockIdx.x * blockDim.x + threadIdx.x;
  int total = B * L * di;
  if (i >= total) return;
  int c = i % di;
  int l = (i / di) % L;
  int b = i / (di * L);
  float acc = conv_b[c];
  for (int k = 0; k < K; ++k) {
    int ls = l - (K - 1) + k;
    if (ls >= 0)
      acc += conv_w[c * K + k] * xz[((size_t)b * L + ls) * (size_t)(2 * di) + c];
  }
  xs_act[i] = silu_f(acc);
}

// ---------------------------------------------------------------------------
// Selective scan.  One thread per (batch, channel): 8192 threads, state h[16]
// in registers, L=256 sequential steps.  Consecutive lanes share (b,l) so the
// Bm/Cm loads are wave-uniform broadcasts and xs/delta loads are coalesced.
// Fuses: softplus(dt@dtw + dt_b), dA=exp(delta*A), h-update, C-contraction,
// +Dp*xs, *silu(z).
// ---------------------------------------------------------------------------
__global__ __launch_bounds__(256)
void mamba_scan(const float* __restrict__ xz,       // for z (row stride 2*di)
                const float* __restrict__ xs_act,   // row stride di
                const float* __restrict__ xdbl,     // row stride 48
                const float* __restrict__ delta_pre,// row stride di
                const float* __restrict__ dt_b,
                const float* __restrict__ Alog,
                const float* __restrict__ Dp,
                float* __restrict__ ygated,         // row stride di
                int B, int L, int di) {
  int t = blockIdx.x * blockDim.x + threadIdx.x;
  if (t >= B * di) return;
  const int c = t % di;
  const int b = t / di;

  float A[DSTATE], h[DSTATE];
  #pragma unroll
  for (int s = 0; s < DSTATE; ++s) {
    A[s] = -__expf(Alog[c * DSTATE + s]);
    h[s] = 0.f;
  }
  const float dpc = Dp[c];
  const float dtb = dt_b[c];

  for (int l = 0; l < L; ++l) {
    const size_t row = (size_t)b * L + l;
    if (l + 1 < L)
      __builtin_prefetch(xs_act + (row + 1) * di + c, 0, 1);

    float dp = delta_pre[row * di + c] + dtb;
    float delta = (dp > 20.f) ? dp : log1pf(__expf(dp));  // softplus
    float xv = xs_act[row * di + c];
    const float* bm = xdbl + row * XDBL_LD + DTR;
    const float* cm = xdbl + row * XDBL_LD + DTR + DSTATE;

    float y = 0.f;
    #pragma unroll
    for (int s = 0; s < DSTATE; ++s) {
      float dA = __expf(delta * A[s]);
      h[s] = dA * h[s] + delta * bm[s] * xv;
      y += h[s] * cm[s];
    }
    float zv = xz[row * (size_t)(2 * di) + di + c];
    ygated[row * di + c] = (y + dpc * xv) * silu_f(zv);
  }
}

// ---------------------------------------------------------------------------
// y0 = LayerNorm(yf + flip(yb) + x).  One block per (b,l) row; LDS reduction.
// ---------------------------------------------------------------------------
__global__ __launch_bounds__(256)
void fuse_add_ln(const float* __restrict__ yf, const float* __restrict__ yb,
                 const float* __restrict__ x,  const float* __restrict__ g,
                 const float* __restrict__ be, float* __restrict__ y0,
                 int L, int d) {
  __shared__ float red[256];
  const int row = blockIdx.x;
  const int b = row / L, l = row % L;
  const int c = threadIdx.x;
  const size_t i  = (size_t)row * d + c;
  const size_t ib = ((size_t)b * L + (L - 1 - l)) * d + c;
  float v = yf[i] + yb[ib] + x[i];

  red[c] = v; __syncthreads();
  for (int o = 128; o > 0; o >>= 1) { if (c < o) red[c] += red[c + o]; __syncthreads(); }
  const float mean = red[0] * (1.f / d);
  __syncthreads();
  const float dv = v - mean;
  red[c] = dv * dv; __syncthreads();
  for (int o = 128; o > 0; o >>= 1) { if (c < o) red[c] += red[c + o]; __syncthreads(); }
  const float var = red[0] * (1.f / d);
  y0[i] = dv * rsqrtf(var + 1e-5f) * g[c] + be[c];
}

// ---------------------------------------------------------------------------
// out = LayerNorm(y1 * y0 + x)
// ---------------------------------------------------------------------------
__global__ __launch_bounds__(256)
void fuse_mul_ln(const float* __restrict__ y1, const float* __restrict__ y0,
                 const float* __restrict__ x,  const float* __restrict__ g,
                 const float* __restrict__ be, float* __restrict__ out,
                 int d) {
  __shared__ float red[256];
  const int row = blockIdx.x;
  const int c = threadIdx.x;
  const size_t i = (size_t)row * d + c;
  float v = y1[i] * y0[i] + x[i];

  red[c] = v; __syncthreads();
  for (int o = 128; o > 0; o >>= 1) { if (c < o) red[c] += red[c + o]; __syncthreads(); }
  const float mean = red[0] * (1.f / d);
  __syncthreads();
  const float dv = v - mean;
  red[c] = dv * dv; __syncthreads();
  for (int o = 128; o > 0; o >>= 1) { if (c < o) red[c] += red[c + o]; __syncthreads(); }
  const float var = red[0] * (1.f / d);
  out[i] = dv * rsqrtf(var + 1e-5f) * g[c] + be[c];
}

// ---------------------------------------------------------------------------
// Per-batch n x n transpose through LDS (32x33 tile, conflict-free).
// ---------------------------------------------------------------------------
__global__ void transpose_b(const float* __restrict__ in, float* __restrict__ out, int n) {
  __shared__ float tile[32][33];
  const int b  = blockIdx.z;
  const int j0 = blockIdx.x * 32, i0 = blockIdx.y * 32;
  const float* src = in  + (size_t)b * n * n;
  float*       dst = out + (size_t)b * n * n;
  for (int r = threadIdx.y; r < 32; r += 8)
    tile[r][threadIdx.x] = src[(size_t)(i0 + r) * n + j0 + threadIdx.x];
  __syncthreads();
  for (int r = threadIdx.y; r < 32; r += 8)
    dst[(size_t)(j0 + r) * n + i0 + threadIdx.x] = tile[threadIdx.x][r];
}

// ---------------------------------------------------------------------------
// Host-side: one full Mamba block over a (B, L, 256) fp32 buffer.
// ---------------------------------------------------------------------------
static void run_mamba(const float* X,
                      const float* in_w, const float* conv_w, const float* conv_b,
                      const float* xproj_w, const float* dt_w, const float* dt_b,
                      const float* Alog, const float* Dp, const float* out_w,
                      float* Y, int convK,
                      float* XZ, float* XS, float* XDBL, float* DELTA, float* YG,
                      hipStream_t stream) {
  const dim3 blk(256);
  // xz = X @ in_w^T        (4096 x 1024, K=256)
  wmma_gemm_xwt<<<dim3((2*DI)/128, MROWS/64), blk, 0, stream>>>(
      X, DMODEL, in_w, DMODEL, XZ, 2*DI, MROWS, 2*DI, DMODEL);
  // depthwise conv + SiLU
  conv_silu<<<(MROWS*DI + 255)/256, blk, 0, stream>>>(
      XZ, conv_w, conv_b, XS, BATCH, SEQ, DI, convK);
  // x_dbl = xs @ xproj_w^T (4096 x 48, K=512)
  wmma_gemm_xwt<<<dim3((XDBL_LD + 127)/128, MROWS/64), blk, 0, stream>>>(
      XS, DI, xproj_w, DI, XDBL, XDBL_LD, MROWS, XDBL_LD, DI);
  // delta_pre = dt @ dt_w^T (4096 x 512, K=16); dt = x_dbl[:, :16] via lda=48
  wmma_gemm_xwt<<<dim3(DI/128, MROWS/64), blk, 0, stream>>>(
      XDBL, XDBL_LD, dt_w, DTR, DELTA, DI, MROWS, DI, DTR);
  // selective scan + gating
  mamba_scan<<<(BATCH*DI + 255)/256, blk, 0, stream>>>(
      XZ, XS, XDBL, DELTA, dt_b, Alog, Dp, YG, BATCH, SEQ, DI);
  // Y = ygated @ out_w^T   (4096 x 256, K=512)
  wmma_gemm_xwt<<<dim3(DMODEL/128, MROWS/64), blk, 0, stream>>>(
      YG, DI, out_w, DI, Y, DMODEL, MROWS, DMODEL, DI);
}

extern "C" void kernel_launch(void* const* d_in, const int* in_sizes, int n_in,
                              void* d_out, int out_size, void* d_ws, size_t ws_size,
                              hipStream_t stream) {
  const float* x    = (const float*)d_in[0];
  const float* ln_g = (const float*)d_in[1];
  const float* ln_b = (const float*)d_in[2];
  // Mamba parameter bundles: [in_w, conv_w, conv_b, xproj_w, dt_w, dt_b, Alog, Dp, out_w]
  const float** mf = (const float**)&d_in[3];
  const float** mb = (const float**)&d_in[12];
  const float** tm = (const float**)&d_in[21];
  float* out = (float*)d_out;

  float* ws = (float*)d_ws;
  size_t off = 0;
  float* XZ    = ws + off; off += (size_t)MROWS * 2 * DI;     // 4,194,304
  float* XS    = ws + off; off += (size_t)MROWS * DI;         // 2,097,152
  float* XDBL  = ws + off; off += (size_t)MROWS * XDBL_LD;    //   196,608
  float* DELTA = ws + off; off += (size_t)MROWS * DI;         // 2,097,152
  float* YG    = ws + off; off += (size_t)MROWS * DI;         // 2,097,152
  float* YF    = ws + off; off += (size_t)MROWS * DMODEL;     // 1,048,576
  float* YB    = ws + off; off += (size_t)MROWS * DMODEL;
  float* XB    = ws + off; off += (size_t)MROWS * DMODEL;     // flipped x, later y1
  float* Y0    = ws + off; off += (size_t)MROWS * DMODEL;
  float* T0    = ws + off; off += (size_t)MROWS * DMODEL;
  float* T1    = ws + off; off += (size_t)MROWS * DMODEL;
  (void)ws_size; (void)n_in; (void)in_sizes; (void)out_size;

  const dim3 blk(256);

  // Backward-branch input: flip x along sequence.
  flip_seq<<<(MROWS*DMODEL + 255)/256, blk, 0, stream>>>(x, XB, BATCH, SEQ, DMODEL);

  // Forward and backward Mamba blocks over the sequence axis.
  run_mamba(x,  mf[0], mf[1], mf[2], mf[3], mf[4], mf[5], mf[6], mf[7], mf[8],
            YF, /*convK=*/1, XZ, XS, XDBL, DELTA, YG, stream);
  run_mamba(XB, mb[0], mb[1], mb[2], mb[3], mb[4], mb[5], mb[6], mb[7], mb[8],
            YB, /*convK=*/1, XZ, XS, XDBL, DELTA, YG, stream);

  // y0 = LN(yf + flip(yb) + x)   (output flip folded into the fusion)
  fuse_add_ln<<<MROWS, blk, 0, stream>>>(YF, YB, x, ln_g, ln_b, Y0, SEQ, DMODEL);

  // Temporal Mamba over the feature axis: transpose, mamba (conv width 4), transpose back.
  transpose_b<<<dim3(8, 8, BATCH), dim3(32, 8), 0, stream>>>(Y0, T0, DMODEL);
  run_mamba(T0, tm[0], tm[1], tm[2], tm[3], tm[4], tm[5], tm[6], tm[7], tm[8],
            T1, /*convK=*/4, XZ, XS, XDBL, DELTA, YG, stream);
  transpose_b<<<dim3(8, 8, BATCH), dim3(32, 8), 0, stream>>>(T1, XB, DMODEL); // XB := y1

  // out = LN(y1 * y0 + x)
  fuse_mul_ln<<<MROWS, blk, 0, stream>>>(XB, Y0, x, ln_g, ln_b, out, DMODEL);
}